// GCNLayer_30116310679884
// MI455X (gfx1250) — compile-verified
//
#include <hip/hip_runtime.h>

typedef float v2f __attribute__((ext_vector_type(2)));
typedef float v8f __attribute__((ext_vector_type(8)));

#define D 128
#define LDS_STRIDE 132  // 128 + 4 pad -> conflict-free ds_load_b64 across 32 lanes

// ---------------------------------------------------------------- init
__global__ void gcn_init_kernel(float* __restrict__ out, float* __restrict__ deg,
                                int n, int total4) {
    int i = blockIdx.x * blockDim.x + threadIdx.x;
    if (i < total4) reinterpret_cast<float4*>(out)[i] = make_float4(0.f, 0.f, 0.f, 0.f);
    if (i < n) deg[i] = 1.0f;  // self-loop contributes 1 to every node's degree
}

// ---------------------------------------------------------------- degree
__global__ void gcn_degree_kernel(const int* __restrict__ rows, float* __restrict__ deg,
                                  int ne) {
    int e = blockIdx.x * blockDim.x + threadIdx.x;
    if (e < ne) atomicAdd(&deg[rows[e]], 1.0f);
}

__global__ void gcn_rsqrt_kernel(float* __restrict__ deg, int n) {
    int i = blockIdx.x * blockDim.x + threadIdx.x;
    if (i < n) deg[i] = rsqrtf(deg[i]);  // deg >= 1 always (self loops)
}

// ---------------------------------------------------------------- GEMM: Hl = H*W^T + b
// One wave = one 16-row tile, 8 col-tiles of 16 => full 128-wide row band.
// A (16x4 f32): lane l<16 -> H[row0+l][k0+{0,1}], lane 16+l -> H[row0+l][k0+{2,3}]
// B (4x16 f32): lane l<16 -> W[nt*16+l][k0+{0,1}], lane 16+l -> W[nt*16+l][k0+{2,3}]
// C/D (16x16):  vgpr v, lane l<16 -> D[v][l];  lane 16+l -> D[v+8][l]
__global__ void __launch_bounds__(256)
gcn_gemm_kernel(const float* __restrict__ H, const float* __restrict__ W,
                const float* __restrict__ bias, float* __restrict__ Hl, int n) {
    __shared__ float Ws[D * LDS_STRIDE];

    const int tid = threadIdx.x;
    // stage W (128x128 = 64KB) into padded LDS, float4 at a time
    for (int idx = tid; idx < (D * D) / 4; idx += 256) {
        int r  = idx >> 5;          // 32 float4 per row
        int c4 = (idx & 31) << 2;
        float4 v = reinterpret_cast<const float4*>(W)[idx];
        float* dst = &Ws[r * LDS_STRIDE + c4];
        dst[0] = v.x; dst[1] = v.y; dst[2] = v.z; dst[3] = v.w;
    }
    __syncthreads();

    const int lane = tid & 31;
    const int wave = tid >> 5;
    const int m    = lane & 15;
    const int kh   = lane >> 4;          // which K-half of the 4-wide slab
    const int numTiles = (n + 15) >> 4;
    const int tile = blockIdx.x * 8 + wave;
    if (tile >= numTiles) return;        // uniform per-wave exit; EXEC stays all-1s

    const int row0 = tile << 4;
    int arow = row0 + m;
    if (arow >= n) arow = n - 1;         // clamp (stores are guarded below)
    const float* hp = H + (long long)arow * D + 2 * kh;

    v8f acc[8];
    const v8f zero = {0.f, 0.f, 0.f, 0.f, 0.f, 0.f, 0.f, 0.f};
#pragma unroll
    for (int t = 0; t < 8; ++t) acc[t] = zero;

    for (int k0 = 0; k0 < D; k0 += 4) {
        v2f a = *reinterpret_cast<const v2f*>(hp + k0);
#pragma unroll
        for (int t = 0; t < 8; ++t) {
            v2f bb = *reinterpret_cast<const v2f*>(&Ws[(t * 16 + m) * LDS_STRIDE + k0 + 2 * kh]);
            acc[t] = __builtin_amdgcn_wmma_f32_16x16x4_f32(
                false, a, false, bb, (short)0, acc[t], false, false);
        }
    }

#pragma unroll
    for (int t = 0; t < 8; ++t) {
        const float bn = bias[t * 16 + m];
#pragma unroll
        for (int v = 0; v < 8; ++v) {
            int row = row0 + v + 8 * kh;
            if (row < n)
                Hl[(long long)row * D + t * 16 + m] = acc[t][v] + bn;
        }
    }
}

// ---------------------------------------------------------------- scatter: out[row] += Hl[col]*norm
// 32 lanes per edge; each lane owns a float4 of the 128-wide row.
__global__ void gcn_scatter_kernel(const float* __restrict__ Hl,
                                   const int* __restrict__ rows,
                                   const int* __restrict__ cols,
                                   const float* __restrict__ dinv,
                                   float* __restrict__ out, int ne, int n) {
    long long t = (long long)blockIdx.x * blockDim.x + threadIdx.x;
    long long e = t >> 5;
    int c4 = (int)(t & 31) << 2;
    if (e >= (long long)ne + n) return;
    int r, c;
    if (e < ne) { r = rows[e]; c = cols[e]; }
    else        { r = c = (int)(e - ne); }  // self loops
    float nm = dinv[r] * dinv[c];
    float4 v = *reinterpret_cast<const float4*>(Hl + (long long)c * D + c4);
    float* o = out + (long long)r * D + c4;
    atomicAdd(o + 0, v.x * nm);
    atomicAdd(o + 1, v.y * nm);
    atomicAdd(o + 2, v.z * nm);
    atomicAdd(o + 3, v.w * nm);
}

// ---------------------------------------------------------------- relu
__global__ void gcn_relu_kernel(float* __restrict__ out, int total4) {
    int i = blockIdx.x * blockDim.x + threadIdx.x;
    if (i < total4) {
        float4 v = reinterpret_cast<float4*>(out)[i];
        v.x = fmaxf(v.x, 0.f); v.y = fmaxf(v.y, 0.f);
        v.z = fmaxf(v.z, 0.f); v.w = fmaxf(v.w, 0.f);
        reinterpret_cast<float4*>(out)[i] = v;
    }
}

// ---------------------------------------------------------------- launch
extern "C" void kernel_launch(void* const* d_in, const int* in_sizes, int n_in,
                              void* d_out, int out_size, void* d_ws, size_t ws_size,
                              hipStream_t stream) {
    const float* H   = (const float*)d_in[0];
    const int*   ei  = (const int*)d_in[1];   // edge_index flattened (2,E) row-major
    const float* W   = (const float*)d_in[2];
    const float* b   = (const float*)d_in[3];
    float*       out = (float*)d_out;

    const int n  = in_sizes[0] / D;
    const int ne = in_sizes[1] / 2;
    const int* rows = ei;
    const int* cols = ei + ne;

    float* deg = (float*)d_ws;                       // n floats -> becomes deg^{-1/2}
    float* Hl  = (float*)d_ws + ((n + 3) & ~3);      // 16B aligned; n*128 floats

    const int total4 = n * (D / 4);

    gcn_init_kernel<<<(total4 + 255) / 256, 256, 0, stream>>>(out, deg, n, total4);
    gcn_degree_kernel<<<(ne + 255) / 256, 256, 0, stream>>>(rows, deg, ne);
    gcn_rsqrt_kernel<<<(n + 255) / 256, 256, 0, stream>>>(deg, n);

    const int numTiles = (n + 15) / 16;
    gcn_gemm_kernel<<<(numTiles + 7) / 8, 256, 0, stream>>>(H, W, b, Hl, n);

    const long long sthreads = ((long long)ne + n) * 32;
    gcn_scatter_kernel<<<(int)((sthreads + 255) / 256), 256, 0, stream>>>(
        Hl, rows, cols, deg, out, ne, n);

    gcn_relu_kernel<<<(total4 + 255) / 256, 256, 0, stream>>>(out, total4);
}